// TreeEncoderNet_79568564126522
// MI455X (gfx1250) — compile-verified
//
#include <hip/hip_runtime.h>
#include <math.h>

#define BB 16
#define LEVELS 12
#define NN 4095
#define HH 256
#define EE 256
#define KC 32

typedef float v2f __attribute__((ext_vector_type(2)));
typedef float v8f __attribute__((ext_vector_type(8)));

__device__ __forceinline__ float sigf(float x) { return 1.0f / (1.0f + expf(-x)); }

// ---------------------------------------------------------------------------
// Generic fp32 WMMA GEMM:  C[row, n] (+)= A_row(row)[k] * Wt[n][wKOff + k]
//   A addressing:
//     if idx:  rowbase = idx[bidx*idxStride + idxBase + inner] * aRowSize
//     else  :  rowbase = bidx*aStrideB + (inner >> rep) * aStrideM
//   with bidx = row / rowsInner, inner = row % rowsInner.
//   Wt is column-major of the original weight (Wt[n*ldw + k] == W[k][n]).
// Block: 256 threads (8 waves). Tile: 32 rows x 256 cols.
//   waves 0-3 -> rows [0:16), waves 4-7 -> rows [16:32); each wave 64 cols.
// A and B staged in double-buffered LDS ([idx][k] layout, stride 34,
// conflict-free b64 fragment reads); next chunk prefetched into registers
// while WMMAs run on the current chunk.
// Requires: kLen % 32 == 0, cols % 256 == 0, rows % 16 == 0 (32-row blocks
// clamp A rows and mask epilogue stores for the tail).
// ---------------------------------------------------------------------------
struct GemmArgs {
  const float* A;
  const int* idx;
  int idxStride;
  int idxBase;
  long long aStrideB;
  long long aStrideM;
  int aRowSize;
  int rowsInner;
  int rep;
  int rows;
  const float* Wt;
  int ldw;
  int wKOff;
  int kLen;
  float* C;
  int ldc;
  int cols;
  int acc;
};

__global__ __launch_bounds__(256) void gemm_f32_wmma(GemmArgs g) {
  __shared__ float sA[2][32 * 34];
  __shared__ float sB[2][256 * 34];

  const int t = threadIdx.x;
  const int lane = t & 31;
  const int wave = t >> 5;
  const int laneLo = lane & 15;
  const int laneHi = lane >> 4;
  const int rowHalf = wave >> 2;      // 0/1: which 16-row half this wave computes
  const int colw = (wave & 3) * 64;   // col offset within block tile
  const int rowTile = blockIdx.y;
  const int colBase0 = blockIdx.x * 256;

  // ---- A global->LDS mapping: thread t loads row (t>>3), 4 floats at (t&7)*4
  const int aRow = t >> 3;            // 0..31
  const int aK = (t & 7) << 2;
  int r = rowTile * 32 + aRow;
  if (r > g.rows - 1) r = g.rows - 1; // clamp (stores masked in epilogue)
  const int bidx = r / g.rowsInner;
  const int inner = r - bidx * g.rowsInner;
  long long aBase;
  if (g.idx) {
    aBase = (long long)g.idx[(long long)bidx * g.idxStride + g.idxBase + inner] *
            (long long)g.aRowSize;
  } else {
    aBase = (long long)bidx * g.aStrideB +
            (long long)(inner >> g.rep) * g.aStrideM;
  }
  const float* Ap = g.A + aBase;

  // ---- B global->LDS mapping: thread t covers n = (t>>3)+32*it, k = (t&7)*4
  const int bRow0 = t >> 3;
  const int bK = (t & 7) << 2;
  const long long ldw = g.ldw;
  const float* WbT = g.Wt + g.wKOff + (long long)(colBase0 + bRow0) * ldw + bK;

  // ---- prologue: load chunk 0 into registers
  float4 aReg = *(const float4*)(Ap + aK);
  float4 bReg[8];
#pragma unroll
  for (int it = 0; it < 8; ++it)
    bReg[it] = *(const float4*)(WbT + 32ll * it * ldw);

  v8f c0 = {}, c1 = {}, c2 = {}, c3 = {};
  int cur = 0;

  for (int kc = 0; kc < g.kLen; kc += KC) {
    // store staged chunk to LDS
    {
      float* sp = &sA[cur][aRow * 34 + aK];
      sp[0] = aReg.x; sp[1] = aReg.y; sp[2] = aReg.z; sp[3] = aReg.w;
#pragma unroll
      for (int it = 0; it < 8; ++it) {
        float* bp = &sB[cur][(bRow0 + 32 * it) * 34 + bK];
        bp[0] = bReg[it].x; bp[1] = bReg[it].y;
        bp[2] = bReg[it].z; bp[3] = bReg[it].w;
      }
    }
    __syncthreads();

    // prefetch next chunk while WMMAs run
    const int kn = kc + KC;
    if (kn < g.kLen) {
      aReg = *(const float4*)(Ap + kn + aK);
#pragma unroll
      for (int it = 0; it < 8; ++it)
        bReg[it] = *(const float4*)(WbT + kn + 32ll * it * ldw);
    }

    const float* sAc = &sA[cur][(rowHalf * 16 + laneLo) * 34 + (laneHi << 1)];
    const float* sBc = &sB[cur][(colw + laneLo) * 34 + (laneHi << 1)];
#pragma unroll
    for (int ks = 0; ks < KC; ks += 4) {
      v2f a;
      a.x = sAc[ks]; a.y = sAc[ks + 1];
      v2f b;
      const float* b0 = sBc + ks;
      b.x = b0[0]; b.y = b0[1];
      c0 = __builtin_amdgcn_wmma_f32_16x16x4_f32(false, a, false, b, (short)0, c0, false, false);
      const float* b1 = b0 + 16 * 34;
      b.x = b1[0]; b.y = b1[1];
      c1 = __builtin_amdgcn_wmma_f32_16x16x4_f32(false, a, false, b, (short)0, c1, false, false);
      const float* b2 = b0 + 32 * 34;
      b.x = b2[0]; b.y = b2[1];
      c2 = __builtin_amdgcn_wmma_f32_16x16x4_f32(false, a, false, b, (short)0, c2, false, false);
      const float* b3 = b0 + 48 * 34;
      b.x = b3[0]; b.y = b3[1];
      c3 = __builtin_amdgcn_wmma_f32_16x16x4_f32(false, a, false, b, (short)0, c3, false, false);
    }
    cur ^= 1;
  }

  // Epilogue. D layout: VGPR i -> row (laneHi*8 + i), col laneLo within tile.
  const int rowB = rowTile * 32 + rowHalf * 16 + (laneHi << 3);
  const int n0 = colBase0 + colw + laneLo;
#pragma unroll
  for (int i = 0; i < 8; i++) {
    const int rr = rowB + i;
    if (rr < g.rows) {
      long long ro = (long long)rr * g.ldc;
      if (g.acc) {
        g.C[ro + n0 + 0]  += c0[i];
        g.C[ro + n0 + 16] += c1[i];
        g.C[ro + n0 + 32] += c2[i];
        g.C[ro + n0 + 48] += c3[i];
      } else {
        g.C[ro + n0 + 0]  = c0[i];
        g.C[ro + n0 + 16] = c1[i];
        g.C[ro + n0 + 32] = c2[i];
        g.C[ro + n0 + 48] = c3[i];
      }
    }
  }
}

// ---------------------------------------------------------------------------
// Weight pack/transpose: dst[(nOff+n)*dstLd + kOff + k] = src[k*srcLd + n]
// ---------------------------------------------------------------------------
__global__ void transpose_pack(const float* src, int srcLd, float* dst, int dstLd,
                               int nOff, int kOff, int K, int Ncols) {
  long long i = (long long)blockIdx.x * 256 + threadIdx.x;
  if (i >= (long long)K * Ncols) return;
  int k = (int)(i / Ncols);
  int n = (int)(i - (long long)k * Ncols);
  dst[(long long)(nOff + n) * dstLd + kOff + k] = src[(long long)k * srcLd + n];
}

// ---------------------------------------------------------------------------
// Bottom-up tree-LSTM cell.
// bigproj (row x 1280): [0:768]=iou, [768:1024]=fproj_child0, [1024:1280]=fproj_child1
// chc = bu_ce + s2*H (children cell states) or nullptr at leaves.
// ---------------------------------------------------------------------------
__global__ __launch_bounds__(256) void bu_cell_kernel(
    const float* bigproj, const float* fxbuf, const float* bioux, const float* bfx,
    const float* chc, float* h_buf, float* c_buf, float* out, int M, int s) {
  long long r = (long long)blockIdx.x * 256 + threadIdx.x;
  int j = (int)(r & (HH - 1));
  long long row = r >> 8;
  int b = (int)(row / M);
  int m = (int)(row - (long long)b * M);
  const float* bp = bigproj + row * 1280;
  float gi = sigf(bp[j] + bioux[j]);
  float go = sigf(bp[HH + j] + bioux[HH + j]);
  float gu = tanhf(bp[2 * HH + j] + bioux[2 * HH + j]);
  float c = gi * gu;
  if (chc) {
    float fx = fxbuf[row * HH + j] + bfx[j];
    long long co = ((long long)b * NN + 2 * m) * HH + j;
    c += sigf(fx + bp[3 * HH + j]) * chc[co];
    c += sigf(fx + bp[4 * HH + j]) * chc[co + HH];
  }
  float h = go * tanhf(c);
  h_buf[row * HH + j] = h;
  c_buf[row * HH + j] = c;
  out[((long long)b * NN + s + m) * (2 * HH) + j] = h;
}

// Edge LSTM cell (BU): writes he/ce at node slots into bu_he/bu_ce.
__global__ __launch_bounds__(256) void edge_cell_bu(
    const float* ep, const float* bih, const float* bhh, const float* c_buf,
    float* he_out, float* ce_out, int M, int s) {
  long long r = (long long)blockIdx.x * 256 + threadIdx.x;
  int j = (int)(r & (HH - 1));
  long long row = r >> 8;
  int b = (int)(row / M);
  int m = (int)(row - (long long)b * M);
  const float* g = ep + row * 1024;
  float gi = g[j] + bih[j] + bhh[j];
  float gf = g[HH + j] + bih[HH + j] + bhh[HH + j];
  float gg = g[2 * HH + j] + bih[2 * HH + j] + bhh[2 * HH + j];
  float go = g[3 * HH + j] + bih[3 * HH + j] + bhh[3 * HH + j];
  float c2 = sigf(gf) * c_buf[row * HH + j] + sigf(gi) * tanhf(gg);
  long long no = ((long long)b * NN + s + m) * HH + j;
  he_out[no] = sigf(go) * tanhf(c2);
  ce_out[no] = c2;
}

// ---------------------------------------------------------------------------
// Top-down cell.
// bigproj (row x 1536): [0:768]=iou(ph+cat parts), [768:1024]=f_ch0, [1024:1280]=f_ch1,
//                       [1280:1536]=f_parent   (x contributions accumulated into [0:768])
// ---------------------------------------------------------------------------
__global__ __launch_bounds__(256) void td_cell_kernel(
    const float* bigproj, const float* fxbuf, const float* bioux, const float* bfx,
    const float* chc, const float* pc, float* h_buf, float* c_buf, float* out,
    int M, int s) {
  long long r = (long long)blockIdx.x * 256 + threadIdx.x;
  int j = (int)(r & (HH - 1));
  long long row = r >> 8;
  int b = (int)(row / M);
  int m = (int)(row - (long long)b * M);
  const float* bp = bigproj + row * 1536;
  float gi = sigf(bp[j] + bioux[j]);
  float go = sigf(bp[HH + j] + bioux[HH + j]);
  float gu = tanhf(bp[2 * HH + j] + bioux[2 * HH + j]);
  float fx = fxbuf[row * HH + j] + bfx[j];
  float c = gi * gu;
  if (chc) {
    long long co = ((long long)b * NN + 2 * m) * HH + j;
    c += sigf(fx + bp[3 * HH + j]) * chc[co];
    c += sigf(fx + bp[4 * HH + j]) * chc[co + HH];
  }
  c += sigf(fx + bp[5 * HH + j]) * pc[row * HH + j];
  float h = go * tanhf(c);
  h_buf[row * HH + j] = h;
  c_buf[row * HH + j] = c;
  out[((long long)b * NN + s + m) * (2 * HH) + HH + j] = h;
}

// Edge LSTM cell (TD): parent states repeated x2 over children; writes ph/pc.
__global__ __launch_bounds__(256) void edge_cell_td(
    const float* ep, const float* bih, const float* bhh, const float* c_buf,
    float* ph, float* pc, int M2) {
  long long r = (long long)blockIdx.x * 256 + threadIdx.x;
  int j = (int)(r & (HH - 1));
  long long row = r >> 8;
  int b = (int)(row / M2);
  int m2 = (int)(row - (long long)b * M2);
  const float* g = ep + row * 1024;
  float gi = g[j] + bih[j] + bhh[j];
  float gf = g[HH + j] + bih[HH + j] + bhh[HH + j];
  float gg = g[2 * HH + j] + bih[2 * HH + j] + bhh[2 * HH + j];
  float go = g[3 * HH + j] + bih[3 * HH + j] + bhh[3 * HH + j];
  float cin = c_buf[((long long)b * (M2 >> 1) + (m2 >> 1)) * HH + j];
  float c2 = sigf(gf) * cin + sigf(gi) * tanhf(gg);
  ph[row * HH + j] = sigf(go) * tanhf(c2);
  pc[row * HH + j] = c2;
}

// ---------------------------------------------------------------------------
extern "C" void kernel_launch(void* const* d_in, const int* in_sizes, int n_in,
                              void* d_out, int out_size, void* d_ws, size_t ws_size,
                              hipStream_t stream) {
  const float* node_emb  = (const float*)d_in[0];
  const float* edge_emb  = (const float*)d_in[1];
  const float* bu_Wioux  = (const float*)d_in[2];
  const float* bu_bioux  = (const float*)d_in[3];
  const float* bu_Wfx    = (const float*)d_in[4];
  const float* bu_bfx    = (const float*)d_in[5];
  const float* bu_Wiouh  = (const float*)d_in[6];
  const float* bu_Wfh    = (const float*)d_in[7];
  const float* bu_eWih   = (const float*)d_in[8];
  const float* bu_eWhh   = (const float*)d_in[9];
  const float* bu_ebih   = (const float*)d_in[10];
  const float* bu_ebhh   = (const float*)d_in[11];
  const float* td_Wioux  = (const float*)d_in[12];
  const float* td_bioux  = (const float*)d_in[13];
  const float* td_Wfx    = (const float*)d_in[14];
  const float* td_bfx    = (const float*)d_in[15];
  const float* td_Wiouhc = (const float*)d_in[16];
  const float* td_Wiouhp = (const float*)d_in[17];
  const float* td_Wfchild= (const float*)d_in[18];
  const float* td_Wfparent=(const float*)d_in[19];
  const float* td_eWih   = (const float*)d_in[20];
  const float* td_eWhh   = (const float*)d_in[21];
  const float* td_ebih   = (const float*)d_in[22];
  const float* td_ebhh   = (const float*)d_in[23];
  const int*   node_ids  = (const int*)d_in[24];
  const int*   edge_ids  = (const int*)d_in[25];
  float* out = (float*)d_out;

  // -------- workspace layout (floats) --------
  float* w = (float*)d_ws;
  size_t off = 0;
  auto alloc = [&](size_t n) { float* p = w + off; off += n; return p; };
  float* Wt_bu_big = alloc((size_t)1280 * 512);
  float* Wt_bu_x   = alloc((size_t)1024 * 256);
  float* Wt_bu_eih = alloc((size_t)1024 * 256);
  float* Wt_bu_ehh = alloc((size_t)1024 * 256);
  float* Wt_td_big = alloc((size_t)1536 * 768);
  float* Wt_td_x   = alloc((size_t)1024 * 256);
  float* Wt_td_eih = alloc((size_t)1024 * 256);
  float* Wt_td_ehh = alloc((size_t)1024 * 256);
  float* bu_he  = alloc((size_t)BB * NN * HH);
  float* bu_ce  = alloc((size_t)BB * NN * HH);
  float* h_buf  = alloc((size_t)BB * 2048 * HH);
  float* c_buf  = alloc((size_t)BB * 2048 * HH);
  float* ph_buf = alloc((size_t)BB * 2048 * HH);
  float* pc_buf = alloc((size_t)BB * 2048 * HH);
  float* fxbuf  = alloc((size_t)32768 * 256);
  float* bigproj = alloc((size_t)32768 * 1536);
  float* edgeproj = bigproj;  // reused after cell kernel consumes bigproj
  if (off * sizeof(float) > ws_size) return;  // workspace too small: bail safely

  // -------- pack weights (transposed, concatenated) --------
  auto tp = [&](const float* src, int srcLd, float* dst, int dstLd, int nOff,
                int kOff, int K, int Nc) {
    long long tot = (long long)K * Nc;
    transpose_pack<<<(unsigned)((tot + 255) / 256), 256, 0, stream>>>(
        src, srcLd, dst, dstLd, nOff, kOff, K, Nc);
  };
  // BU big: cols [0:768]=Wiouh, [768:1024]=Wfh[0], [1024:1280]=Wfh[1]; K=512
  tp(bu_Wiouh, 768, Wt_bu_big, 512, 0, 0, 512, 768);
  tp(bu_Wfh + 0,         256, Wt_bu_big, 512, 768,  0, 512, 256);
  tp(bu_Wfh + 512 * 256, 256, Wt_bu_big, 512, 1024, 0, 512, 256);
  // BU x: cols [0:768]=Wioux, [768:1024]=Wfx; K=256
  tp(bu_Wioux, 768, Wt_bu_x, 256, 0, 0, 256, 768);
  tp(bu_Wfx,   256, Wt_bu_x, 256, 768, 0, 256, 256);
  tp(bu_eWih, 1024, Wt_bu_eih, 256, 0, 0, 256, 1024);
  tp(bu_eWhh, 1024, Wt_bu_ehh, 256, 0, 0, 256, 1024);
  // TD big (K=768, pcat=[ph(256) | cat(512)]):
  //   cols [0:768]: k<256 -> Wiouhp, k>=256 -> Wiouhc
  //   cols [768:1024]=Wfchild[0], [1024:1280]=Wfchild[1], [1280:1536]=Wfparent
  tp(td_Wiouhp, 768, Wt_td_big, 768, 0, 0,   256, 768);
  tp(td_Wiouhc, 768, Wt_td_big, 768, 0, 256, 512, 768);
  tp(td_Wfchild + 0,         256, Wt_td_big, 768, 768,  0, 768, 256);
  tp(td_Wfchild + 768 * 256, 256, Wt_td_big, 768, 1024, 0, 768, 256);
  tp(td_Wfparent, 256, Wt_td_big, 768, 1280, 0, 768, 256);
  tp(td_Wioux, 768, Wt_td_x, 256, 0, 0, 256, 768);
  tp(td_Wfx,   256, Wt_td_x, 256, 768, 0, 256, 256);
  tp(td_eWih, 1024, Wt_td_eih, 256, 0, 0, 256, 1024);
  tp(td_eWhh, 1024, Wt_td_ehh, 256, 0, 0, 256, 1024);

  auto gemm = [&](int rows, const float* A, const int* idx, int idxBase,
                  long long sB, long long sM, int rowsInner, int rep,
                  const float* Wt, int ldw, int wKOff, int kLen, float* C,
                  int ldc, int cols, int acc) {
    GemmArgs ga;
    ga.A = A; ga.idx = idx; ga.idxStride = NN; ga.idxBase = idxBase;
    ga.aStrideB = sB; ga.aStrideM = sM; ga.aRowSize = EE;
    ga.rowsInner = rowsInner; ga.rep = rep; ga.rows = rows;
    ga.Wt = Wt; ga.ldw = ldw; ga.wKOff = wKOff; ga.kLen = kLen;
    ga.C = C; ga.ldc = ldc; ga.cols = cols; ga.acc = acc;
    dim3 grid((unsigned)(cols / 256), (unsigned)((rows + 31) / 32));
    gemm_f32_wmma<<<grid, 256, 0, stream>>>(ga);
  };

  // ================= bottom-up =================
  for (int l = LEVELS - 1; l >= 0; --l) {
    int M = 1 << l, s = M - 1, s2 = 2 * M - 1;
    int R = BB * M;
    bool leaf = (l == LEVELS - 1);
    if (!leaf)  // bigproj = cat @ [Wiouh|Wfh0|Wfh1]
      gemm(R, bu_he + (long long)s2 * HH, nullptr, 0, (long long)NN * HH,
           2 * HH, M, 0, Wt_bu_big, 512, 0, 512, bigproj, 1280, 1280, 0);
    // bigproj[:,0:768] (+)= x @ Wioux
    gemm(R, node_emb, node_ids, s, 0, 0, M, 0, Wt_bu_x, 256, 0, 256, bigproj,
         1280, 768, leaf ? 0 : 1);
    // fxbuf = x @ Wfx
    gemm(R, node_emb, node_ids, s, 0, 0, M, 0, Wt_bu_x + 768ll * 256, 256, 0,
         256, fxbuf, 256, 256, 0);
    bu_cell_kernel<<<(unsigned)R, 256, 0, stream>>>(
        bigproj, fxbuf, bu_bioux, bu_bfx,
        leaf ? nullptr : (bu_ce + (long long)s2 * HH), h_buf, c_buf, out, M, s);
    // edgeproj = e @ eWih ; += h @ eWhh
    gemm(R, edge_emb, edge_ids, s, 0, 0, M, 0, Wt_bu_eih, 256, 0, 256, edgeproj,
         1024, 1024, 0);
    gemm(R, h_buf, nullptr, 0, (long long)M * HH, HH, M, 0, Wt_bu_ehh, 256, 0,
         256, edgeproj, 1024, 1024, 1);
    edge_cell_bu<<<(unsigned)R, 256, 0, stream>>>(edgeproj, bu_ebih, bu_ebhh,
                                                  c_buf, bu_he, bu_ce, M, s);
  }

  // ================= top-down =================
  hipMemsetAsync(ph_buf, 0, (size_t)BB * HH * sizeof(float), stream);
  hipMemsetAsync(pc_buf, 0, (size_t)BB * HH * sizeof(float), stream);
  for (int l = 0; l < LEVELS; ++l) {
    int M = 1 << l, s = M - 1, s2 = 2 * M - 1;
    int R = BB * M;
    bool leaf = (l == LEVELS - 1);
    // bigproj = ph @ Wt_td_big[k 0:256]
    gemm(R, ph_buf, nullptr, 0, (long long)M * HH, HH, M, 0, Wt_td_big, 768, 0,
         256, bigproj, 1536, 1536, 0);
    if (!leaf)  // += cat @ Wt_td_big[k 256:768]
      gemm(R, bu_he + (long long)s2 * HH, nullptr, 0, (long long)NN * HH,
           2 * HH, M, 0, Wt_td_big, 768, 256, 512, bigproj, 1536, 1536, 1);
    // bigproj[:,0:768] += x @ Wioux
    gemm(R, node_emb, node_ids, s, 0, 0, M, 0, Wt_td_x, 256, 0, 256, bigproj,
         1536, 768, 1);
    // fxbuf = x @ Wfx
    gemm(R, node_emb, node_ids, s, 0, 0, M, 0, Wt_td_x + 768ll * 256, 256, 0,
         256, fxbuf, 256, 256, 0);
    td_cell_kernel<<<(unsigned)R, 256, 0, stream>>>(
        bigproj, fxbuf, td_bioux, td_bfx,
        leaf ? nullptr : (bu_ce + (long long)s2 * HH), pc_buf, h_buf, c_buf,
        out, M, s);
    if (!leaf) {
      int M2 = 2 * M, R2 = BB * M2;
      gemm(R2, edge_emb, edge_ids, s2, 0, 0, M2, 0, Wt_td_eih, 256, 0, 256,
           edgeproj, 1024, 1024, 0);
      // repeated parent h: inner>>1
      gemm(R2, h_buf, nullptr, 0, (long long)M * HH, HH, M2, 1, Wt_td_ehh, 256,
           0, 256, edgeproj, 1024, 1024, 1);
      edge_cell_td<<<(unsigned)R2, 256, 0, stream>>>(
          edgeproj, td_ebih, td_ebhh, c_buf, ph_buf, pc_buf, M2);
    }
  }
}